// GIN_4733053960252
// MI455X (gfx1250) — compile-verified
//
#include <hip/hip_runtime.h>

typedef __attribute__((ext_vector_type(2))) float v2f;
typedef __attribute__((ext_vector_type(8))) float v8f;

#define N_NODES  50000
#define N_EDGES  800000
#define N_GRAPHS 128
#define DIM      128
#define OUT_DIM  64
#define N_CONV   4
#define BN_EPS   1e-5f

// ---------------------------------------------------------------------------
// zero a buffer with float4 stores
// ---------------------------------------------------------------------------
__global__ void gin_zero4_kernel(float4* __restrict__ p, unsigned n4) {
    unsigned i = blockIdx.x * blockDim.x + threadIdx.x;
    if (i < n4) p[i] = make_float4(0.f, 0.f, 0.f, 0.f);
}

__global__ void gin_zero_kernel(float* __restrict__ p, int n) {
    int i = blockIdx.x * blockDim.x + threadIdx.x;
    if (i < n) p[i] = 0.0f;
}

// ---------------------------------------------------------------------------
// agg[dst] += h[src]  : 32 threads per edge, 4 floats per thread
// ---------------------------------------------------------------------------
__global__ void gin_scatter_kernel(const float* __restrict__ h,
                                   const int* __restrict__ esrc,
                                   const int* __restrict__ edst,
                                   float* __restrict__ agg) {
    unsigned tid = blockIdx.x * blockDim.x + threadIdx.x;
    unsigned e = tid >> 5;
    if (e >= N_EDGES) return;
    unsigned part = (tid & 31u) * 4u;
    int s = esrc[e];
    int d = edst[e];
    float4 v = *(const float4*)(h + (size_t)s * DIM + part);
    float* o = agg + (size_t)d * DIM + part;
    atomicAdd(o + 0, v.x);
    atomicAdd(o + 1, v.y);
    atomicAdd(o + 2, v.z);
    atomicAdd(o + 3, v.w);
}

// ---------------------------------------------------------------------------
// C[M,128] = (A (+ A2)) @ B[128,128], f32, via V_WMMA_F32_16X16X4_F32.
// One wave per 16-row strip computing all 128 output columns:
//   8 accumulators (v8f each, 64 VGPRs), A loaded once per k-step.
// A 16x4 layout : lanes 0-15 -> K={k0,k0+1}, lanes 16-31 -> K={k0+2,k0+3}
// B 4x16 layout : mirrored; C/D: VGPR r -> row (r + 8*laneHalf), col = lane&15
// Epilogue also accumulates per-column sum / sumsq into stats[0:128]/[128:256]
// for the training-mode batchnorm that follows.
// ---------------------------------------------------------------------------
template <bool ADD2>
__global__ __launch_bounds__(256) void gin_gemm_wmma_kernel(
    const float* __restrict__ A, const float* __restrict__ A2,
    const float* __restrict__ B, float* __restrict__ C,
    float* __restrict__ stats) {
    const int lane  = threadIdx.x & 31;
    const int gwave = (blockIdx.x * blockDim.x + threadIdx.x) >> 5;
    const int mTiles = N_NODES / 16;                 // 3125 strips
    if (gwave >= mTiles) return;                     // wave-uniform exit
    const int half = lane >> 4;                      // 0 or 1
    const int l15  = lane & 15;
    const int kgrp = half * 2;                       // K offset 0 or 2
    const size_t rowOff = (size_t)(gwave * 16 + l15) * DIM;
    const float* Arow  = A + rowOff;
    const float* A2row = A2 + rowOff;

    v8f acc[8];
#pragma unroll
    for (int t = 0; t < 8; ++t) acc[t] = (v8f){};

#pragma unroll 2
    for (int k0 = 0; k0 < DIM; k0 += 4) {
        v2f a = *(const v2f*)(Arow + k0 + kgrp);
        if (ADD2) {
            v2f a2 = *(const v2f*)(A2row + k0 + kgrp);
            a.x += a2.x;
            a.y += a2.y;
        }
        const float* Bp = B + (size_t)(k0 + kgrp) * DIM + l15;
#pragma unroll
        for (int t = 0; t < 8; ++t) {
            v2f b;
            b.x = Bp[t * 16];
            b.y = Bp[t * 16 + DIM];
            acc[t] = __builtin_amdgcn_wmma_f32_16x16x4_f32(
                /*neg_a=*/false, a, /*neg_b=*/false, b,
                /*c_mod=*/(short)0, acc[t], /*reuse_a=*/false, /*reuse_b=*/false);
        }
    }

    float* Cbase = C + (size_t)(gwave * 16 + half * 8) * DIM + l15;
#pragma unroll
    for (int t = 0; t < 8; ++t) {
        float s = 0.f, q = 0.f;
#pragma unroll
        for (int r = 0; r < 8; ++r) {
            float x = acc[t][r];
            Cbase[(size_t)r * DIM + t * 16] = x;
            s += x;
            q += x * x;
        }
        atomicAdd(&stats[t * 16 + l15], s);
        atomicAdd(&stats[DIM + t * 16 + l15], q);
    }
}

// ---------------------------------------------------------------------------
// in-place x = relu((x - mean) * rsqrt(var + eps) * g + b)
// ---------------------------------------------------------------------------
__global__ void gin_bnrelu_kernel(float* __restrict__ Z,
                                  const float* __restrict__ stats,
                                  const float* __restrict__ g,
                                  const float* __restrict__ b) {
    unsigned idx = blockIdx.x * blockDim.x + threadIdx.x;
    const unsigned total = (unsigned)N_NODES * DIM;
    if (idx >= total) return;
    int c = idx & (DIM - 1);
    const float invN = 1.0f / (float)N_NODES;
    float mean = stats[c] * invN;
    float var  = stats[DIM + c] * invN - mean * mean;
    float x = Z[idx];
    x = (x - mean) * rsqrtf(var + BN_EPS) * g[c] + b[c];
    Z[idx] = fmaxf(x, 0.0f);
}

// ---------------------------------------------------------------------------
// BN + ReLU, plus fused graph pooling: pooled[gid[row]][c] += x
// ---------------------------------------------------------------------------
__global__ void gin_bnrelu_pool_kernel(float* __restrict__ Z,
                                       const float* __restrict__ stats,
                                       const float* __restrict__ g,
                                       const float* __restrict__ b,
                                       const int* __restrict__ gid,
                                       float* __restrict__ pooled) {
    unsigned idx = blockIdx.x * blockDim.x + threadIdx.x;
    const unsigned total = (unsigned)N_NODES * DIM;
    if (idx >= total) return;
    int c   = idx & (DIM - 1);
    int row = idx >> 7;
    const float invN = 1.0f / (float)N_NODES;
    float mean = stats[c] * invN;
    float var  = stats[DIM + c] * invN - mean * mean;
    float x = Z[idx];
    x = (x - mean) * rsqrtf(var + BN_EPS) * g[c] + b[c];
    x = fmaxf(x, 0.0f);
    Z[idx] = x;
    atomicAdd(&pooled[(size_t)gid[row] * DIM + c], x);
}

// ---------------------------------------------------------------------------
// pooled[gid[n]] += rep[n]  (used only for the raw input rep h0)
// ---------------------------------------------------------------------------
__global__ void gin_pool_kernel(const float* __restrict__ rep,
                                const int* __restrict__ gid,
                                float* __restrict__ pooled) {
    unsigned tid = blockIdx.x * blockDim.x + threadIdx.x;
    unsigned n = tid >> 5;
    if (n >= N_NODES) return;
    unsigned part = (tid & 31u) * 4u;
    int g = gid[n];
    float4 v = *(const float4*)(rep + (size_t)n * DIM + part);
    float* o = pooled + (size_t)g * DIM + part;
    atomicAdd(o + 0, v.x);
    atomicAdd(o + 1, v.y);
    atomicAdd(o + 2, v.z);
    atomicAdd(o + 3, v.w);
}

// ---------------------------------------------------------------------------
// score[g][o] = sum_r ( pooled[r][g] . predW[r][:,o] + predb[r][o] )
// ---------------------------------------------------------------------------
__global__ void gin_final_kernel(const float* __restrict__ pooled,  // [5][128][128]
                                 const float* __restrict__ predW,   // [5][128][64]
                                 const float* __restrict__ predb,   // [5][64]
                                 float* __restrict__ out) {         // [128][64]
    int g = blockIdx.x;
    int o = threadIdx.x;
    float acc = 0.0f;
#pragma unroll
    for (int r = 0; r < 5; ++r) {
        const float* p = pooled + ((size_t)r * N_GRAPHS + g) * DIM;
        const float* w = predW + (size_t)r * DIM * OUT_DIM + o;
        float a = 0.0f;
        for (int k = 0; k < DIM; ++k) a += p[k] * w[(size_t)k * OUT_DIM];
        acc += a + predb[r * OUT_DIM + o];
    }
    out[(size_t)g * OUT_DIM + o] = acc;
}

// ---------------------------------------------------------------------------
extern "C" void kernel_launch(void* const* d_in, const int* in_sizes, int n_in,
                              void* d_out, int out_size, void* d_ws, size_t ws_size,
                              hipStream_t stream) {
    (void)in_sizes; (void)n_in; (void)out_size; (void)ws_size;

    const float* h0    = (const float*)d_in[0];
    const int*   esrc  = (const int*)d_in[1];
    const int*   edst  = (const int*)d_in[2];
    const int*   gid   = (const int*)d_in[3];
    const float* W1    = (const float*)d_in[4];
    const float* bn1g  = (const float*)d_in[5];
    const float* bn1b  = (const float*)d_in[6];
    const float* W2    = (const float*)d_in[7];
    const float* bn2g  = (const float*)d_in[8];
    const float* bn2b  = (const float*)d_in[9];
    const float* predW = (const float*)d_in[10];
    const float* predb = (const float*)d_in[11];
    float* out = (float*)d_out;

    const size_t featBytes = (size_t)N_NODES * DIM * sizeof(float);
    char* ws = (char*)d_ws;
    float* agg = (float*)ws; ws += featBytes;
    float* Z1  = (float*)ws; ws += featBytes;
    float* hb[N_CONV];
    for (int i = 0; i < N_CONV; ++i) { hb[i] = (float*)ws; ws += featBytes; }
    float* pooled = (float*)ws; ws += (size_t)5 * N_GRAPHS * DIM * sizeof(float);
    float* stats  = (float*)ws;  // 256 floats

    const unsigned feat4      = (unsigned)(N_NODES * DIM / 4);      // 1.6M float4
    const int zero4Blocks     = (int)((feat4 + 255) / 256);         // 6250
    const int scatterBlocks   = (N_EDGES * 32) / 256;               // 100000
    const int gemmBlocks      = (N_NODES / 16 + 7) / 8;             // 391 (8 waves/block)
    const int bnBlocks        = (N_NODES * DIM + 255) / 256;        // 25000
    const int poolBlocks      = (N_NODES * 32 + 255) / 256;         // 6250
    const int pooledElems     = 5 * N_GRAPHS * DIM;

    // pooled must be zero before the fused BN+pool passes run
    gin_zero_kernel<<<(pooledElems + 255) / 256, 256, 0, stream>>>(pooled, pooledElems);

    const float* hcur = h0;
    for (int i = 0; i < N_CONV; ++i) {
        // neighbor sum into zeroed agg; self-term added inside GEMM1's A-load
        gin_zero4_kernel<<<zero4Blocks, 256, 0, stream>>>((float4*)agg, feat4);
        gin_scatter_kernel<<<scatterBlocks, 256, 0, stream>>>(hcur, esrc, edst, agg);

        // Z1 = (agg + hcur) @ W1[i]  (+ fused column stats), then BN + ReLU
        gin_zero_kernel<<<1, 256, 0, stream>>>(stats, 2 * DIM);
        gin_gemm_wmma_kernel<true><<<gemmBlocks, 256, 0, stream>>>(
            agg, hcur, W1 + (size_t)i * DIM * DIM, Z1, stats);
        gin_bnrelu_kernel<<<bnBlocks, 256, 0, stream>>>(Z1, stats, bn1g + i * DIM, bn1b + i * DIM);

        // hb[i] = Z1 @ W2[i]  (+ fused column stats), then BN + ReLU + graph pooling
        gin_zero_kernel<<<1, 256, 0, stream>>>(stats, 2 * DIM);
        gin_gemm_wmma_kernel<false><<<gemmBlocks, 256, 0, stream>>>(
            Z1, Z1, W2 + (size_t)i * DIM * DIM, hb[i], stats);
        gin_bnrelu_pool_kernel<<<bnBlocks, 256, 0, stream>>>(
            hb[i], stats, bn2g + i * DIM, bn2b + i * DIM, gid,
            pooled + (size_t)(i + 1) * N_GRAPHS * DIM);

        hcur = hb[i];
    }

    // pooling of the raw input rep h0
    gin_pool_kernel<<<poolBlocks, 256, 0, stream>>>(h0, gid, pooled);

    // final prediction heads
    gin_final_kernel<<<N_GRAPHS, OUT_DIM, 0, stream>>>(pooled, predW, predb, out);
}